// BaselineParser_45887430590984
// MI455X (gfx1250) — compile-verified
//
#include <hip/hip_runtime.h>
#include <hip/hip_bf16.h>
#include <stdint.h>

// Problem constants (match reference)
#define B_ 32
#define S_ 2048
#define D_ 768
#define W_ 1024
#define TPB 192   // 6 wave32s; 192 lanes * 4 floats = 768 = D

// ---- CDNA5 async global->LDS copy (tracked by ASYNCcnt) ----
// ISA 15.18.3: GLOBAL_LOAD_ASYNC_TO_LDS_B128, GV mode (SADDR=off):
//   vdst = VGPR holding LDS byte address, vaddr = VGPR pair holding 64-bit global address
__device__ __forceinline__ void async_load_b128(unsigned lds_byte_addr, const void* gaddr) {
    asm volatile("global_load_async_to_lds_b128 %0, %1, off"
                 :: "v"(lds_byte_addr), "v"(gaddr)
                 : "memory");
}

#if __has_builtin(__builtin_amdgcn_s_wait_asynccnt)
#define WAIT_ASYNC(n) __builtin_amdgcn_s_wait_asynccnt(n)
#else
#define WAIT_ASYNC(n) asm volatile("s_wait_asynccnt %0" :: "n"(n) : "memory")
#endif

// Uniform lower_bound over one batch row of word_ids (all lanes identical -> coalesced)
__device__ __forceinline__ int lower_bound_dev(const int* __restrict__ a, int lo, int hi, int key) {
    while (lo < hi) {
        int mid = (lo + hi) >> 1;
        if (a[mid] < key) lo = mid + 1; else hi = mid;
    }
    return lo;
}

__global__ __launch_bounds__(TPB)
void meanpool_words_kernel(const float* __restrict__ hidden,   // [B,S,D]
                           const int*   __restrict__ word_ids, // [B,S] sorted per row
                           float* __restrict__ out)            // [B*W*D] X, then [B*W] mask
{
    __shared__ __align__(16) float sbuf[2][TPB * 4];   // two 3072-byte row buffers

    const int gw = blockIdx.x;            // 0 .. B*W-1
    const int b  = gw >> 10;              // / W_
    const int w  = gw & (W_ - 1);         // % W_
    const int t  = threadIdx.x;

    const int* ids = word_ids + b * S_;
    const int lo  = lower_bound_dev(ids, 0, S_, w);
    const int hi  = lower_bound_dev(ids, lo, S_, w + 1);
    const int cnt = hi - lo;

    float4 acc = make_float4(0.f, 0.f, 0.f, 0.f);

    if (cnt > 0) {
        // lane t owns 16 bytes (columns 4t..4t+3) of each row
        const float* rowbase = hidden + ((size_t)b * S_ + (size_t)lo) * D_ + t * 4;
        const unsigned lds0 = (unsigned)(uintptr_t)&sbuf[0][t * 4];
        const unsigned lds1 = (unsigned)(uintptr_t)&sbuf[1][t * 4];

        // prologue: prefetch first row into buffer 0
        async_load_b128(lds0, rowbase);

        for (int r = 0; r < cnt; ++r) {
            const bool more = (r + 1 < cnt);
            if (more) {
                // prefetch next row into the other buffer
                async_load_b128((r & 1) ? lds0 : lds1,
                                rowbase + (size_t)(r + 1) * D_);
                WAIT_ASYNC(1);   // async loads complete in order -> row r is ready
            } else {
                WAIT_ASYNC(0);
            }
            const float4 v = *(const float4*)&sbuf[r & 1][t * 4]; // ds_load_b128
            acc.x += v.x; acc.y += v.y; acc.z += v.z; acc.w += v.w;
            // compiler-inserted s_wait_dscnt before the adds also fences this
            // buffer before it is re-targeted in the next iteration
        }
        const float inv = 1.0f / (float)cnt;
        acc.x *= inv; acc.y *= inv; acc.z *= inv; acc.w *= inv;
    }

    // X output (zeros where cnt==0, since d_out is poisoned)
    float4* outX = (float4*)(out + (size_t)gw * D_);
    outX[t] = acc;

    // mask output, appended after X, as 0.0/1.0
    if (t == 0) {
        out[(size_t)B_ * W_ * D_ + gw] = (cnt > 0) ? 1.0f : 0.0f;
    }
}

extern "C" void kernel_launch(void* const* d_in, const int* in_sizes, int n_in,
                              void* d_out, int out_size, void* d_ws, size_t ws_size,
                              hipStream_t stream) {
    const float* hidden   = (const float*)d_in[0]; // [B,S,D] f32
    const int*   word_ids = (const int*)  d_in[1]; // [B,S] i32
    float*       out      = (float*)d_out;         // [B*W*D + B*W] f32

    (void)in_sizes; (void)n_in; (void)out_size; (void)d_ws; (void)ws_size;

    meanpool_words_kernel<<<dim3(B_ * W_), dim3(TPB), 0, stream>>>(hidden, word_ids, out);
}